// RefineAutoEncoder_214748365316
// MI455X (gfx1250) — compile-verified
//
#include <hip/hip_runtime.h>

typedef __attribute__((ext_vector_type(2))) float v2f;
typedef __attribute__((ext_vector_type(8))) float v8f;

#define TPB 256

// ---- problem constants (match reference) ----
#define B_   32
#define N0_  35709
#define N1_  8928
#define N2_  2232
#define E0_  285672
#define E2_  71424
#define E3_  17856
#define ND0_ (4 * N1_)
#define ND1_ (4 * N2_)
#define NU3_ (3 * N1_)
#define NU2_ (3 * N0_)

// ---------------- elementwise / setup kernels ----------------

__global__ void k_zero(float* __restrict__ p, int n) {
  int i = blockIdx.x * blockDim.x + threadIdx.x;
  if (i < n) p[i] = 0.f;
}

__global__ void k_deg(const int* __restrict__ row, int E, float* __restrict__ deg) {
  int i = blockIdx.x * blockDim.x + threadIdx.x;
  if (i < E) atomicAdd(&deg[row[i]], 1.f);
}

__global__ void k_dinv(float* __restrict__ d, int n) {
  int i = blockIdx.x * blockDim.x + threadIdx.x;
  if (i < n) {
    float v = d[i];
    d[i] = (v > 0.f) ? (1.f / sqrtf(v)) : 0.f;
  }
}

// w[e] = c * dinv[row[e]] * dinv[col[e]],  c = -2/lambda
__global__ void k_edgew(const int* __restrict__ row, const int* __restrict__ col,
                        const float* __restrict__ dinv, float c,
                        float* __restrict__ w, int E) {
  int i = blockIdx.x * blockDim.x + threadIdx.x;
  if (i < E) w[i] = c * dinv[row[i]] * dinv[col[i]];
}

// x[B,N,F] (batch-major input) -> Tstack[:, 0:F) with row stride K, rows = N*B
__global__ void k_t0_from_x(const float* __restrict__ x, float* __restrict__ T,
                            int N, int F, int K, int total) {
  int t = blockIdx.x * blockDim.x + threadIdx.x;
  if (t >= total) return;
  int f = t % F; int t2 = t / F;
  int b = t2 % B_; int nidx = t2 / B_;
  T[(nidx * B_ + b) * K + f] = x[(b * N + nidx) * F + f];
}

// src[rows,F] -> Tstack[:, 0:F) with row stride K
__global__ void k_t0_copy(const float* __restrict__ src, float* __restrict__ T,
                          int F, int K, int total /* rows*F */) {
  int t = blockIdx.x * blockDim.x + threadIdx.x;
  if (t >= total) return;
  int f = t % F; int r = t / F;
  T[r * K + f] = src[r * F + f];
}

// T[:,dOff+f] = alpha*T[:,s1Off+f] + beta*T[:,s2Off+f]   (Chebyshev diagonal/recurrence)
__global__ void k_lincomb(float* __restrict__ T, int K, int dOff, int s1Off, float alpha,
                          int s2Off, float beta, int F, int total /* rows*F */) {
  int t = blockIdx.x * blockDim.x + threadIdx.x;
  if (t >= total) return;
  int f = t % F; int r = t / F;
  int base = r * K;
  T[base + dOff + f] = alpha * T[base + s1Off + f] + beta * T[base + s2Off + f];
}

// scalar scatter (any F):
// dst[(row*B+b)*dstStride+f] += scale*w[e]*src[(col*B+b)*srcStride+f]
__global__ void k_scatter(const int* __restrict__ row, const int* __restrict__ col,
                          const float* __restrict__ w, float scale,
                          const float* __restrict__ src, float* __restrict__ dst,
                          int srcStride, int dstStride, int F, int total /* E*B*F */) {
  int t = blockIdx.x * blockDim.x + threadIdx.x;
  if (t >= total) return;
  int f = t % F; int t2 = t / F;
  int b = t2 % B_; int e = t2 / B_;
  float wv = scale * w[e];
  int si = (col[e] * B_ + b) * srcStride + f;
  int di = (row[e] * B_ + b) * dstStride + f;
  atomicAdd(&dst[di], wv * src[si]);
}

// vectorized scatter (F % 4 == 0): one thread per (e, b, f-quad).
// b128 source load + 4 atomics; amortizes row/col/w index traffic 4x.
__global__ void k_scatter4(const int* __restrict__ row, const int* __restrict__ col,
                           const float* __restrict__ w, float scale,
                           const float* __restrict__ src, float* __restrict__ dst,
                           int srcStride, int dstStride, int F4 /* F/4 */,
                           int total /* E*B*F4 */) {
  int t = blockIdx.x * blockDim.x + threadIdx.x;
  if (t >= total) return;
  int q = t % F4; int t2 = t / F4;
  int b = t2 % B_; int e = t2 / B_;
  float wv = scale * w[e];
  const float4* s =
      (const float4*)(src + (size_t)(col[e] * B_ + b) * srcStride) + q;
  float* d = dst + (size_t)(row[e] * B_ + b) * dstStride + q * 4;
  float4 v = *s;
  atomicAdd(&d[0], wv * v.x);
  atomicAdd(&d[1], wv * v.y);
  atomicAdd(&d[2], wv * v.z);
  atomicAdd(&d[3], wv * v.w);
}

// pack W[K,Fout] into Wp[K,16], zero-padded columns
__global__ void k_packw(const float* __restrict__ Wsrc, float* __restrict__ Wp,
                        int Krows, int Fout) {
  int t = blockIdx.x * blockDim.x + threadIdx.x;
  if (t >= Krows * 16) return;
  int fo = t % 16; int kr = t / 16;
  Wp[t] = (fo < Fout) ? Wsrc[kr * Fout + fo] : 0.f;
}

// ---------------- WMMA GEMM:  out[M,0:Fout) = T[M,K] @ Wp[K,16] + bias ----------------
// One wave32 computes one 16x16 tile with V_WMMA_F32_16X16X4_F32.
// A 16x4 f32 frag: lanes 0-15 -> (M=lane, K=k,k+1); lanes 16-31 -> (M=lane-16, K=k+2,k+3).
// B 4x16 frag: same half-wave K split, N = lane&15.  C/D v8f: VGPR r = rows r and r+8.
// Weights staged once per block in LDS (<= 12 KB of the 320 KB WGP pool).
__global__ __launch_bounds__(TPB)
void k_gemm_wmma(const float* __restrict__ T, int K, const float* __restrict__ Wp,
                 const float* __restrict__ bias, float* __restrict__ out,
                 int M, int Fout, int outStride, int applyElu, int batchMajorN) {
  __shared__ float sW[192 * 16];
  int nW = K * 16;
  for (int i = threadIdx.x; i < nW; i += TPB) sW[i] = Wp[i];
  __syncthreads();                    // before any wave exits (barrier membership)

  int wv = blockIdx.x * (TPB / 32) + (threadIdx.x >> 5);
  int tiles = M >> 4;
  if (wv >= tiles) return;            // wave-uniform exit: EXEC all-ones for WMMA
  int lane = threadIdx.x & 31;
  int hs = lane >> 4;
  int lm = lane & 15;
  int m0 = wv << 4;

  const float* arow = T + (size_t)(m0 + lm) * K + hs * 2;
  const float* brow = sW + hs * 32 + lm;   // row (k + hs*2), col lm
  v8f acc = {0.f, 0.f, 0.f, 0.f, 0.f, 0.f, 0.f, 0.f};

  for (int k = 0; k < K; k += 4) {
    v2f a, b;
    a.x = arow[k];
    a.y = arow[k + 1];
    b.x = brow[k * 16];
    b.y = brow[k * 16 + 16];
    acc = __builtin_amdgcn_wmma_f32_16x16x4_f32(false, a, false, b,
                                                (short)0, acc, false, false);
  }

  if (lm < Fout) {
    float bv = bias[lm];
#pragma unroll
    for (int r = 0; r < 8; r++) {
      int m = m0 + r + hs * 8;
      float v = acc[r] + bv;
      if (applyElu) v = (v > 0.f) ? v : (expf(v) - 1.f);
      int o;
      if (batchMajorN) {              // final layer writes [B, N, Fout] batch-major
        int node = m / B_;
        int bb = m - node * B_;
        o = (bb * batchMajorN + node) * outStride + lm;
      } else {
        o = m * outStride + lm;
      }
      out[o] = v;
    }
  }
}

// ---------------- host driver ----------------

extern "C" void kernel_launch(void* const* d_in, const int* in_sizes, int n_in,
                              void* d_out, int out_size, void* d_ws, size_t ws_size,
                              hipStream_t stream) {
  (void)in_sizes; (void)n_in; (void)out_size; (void)ws_size;

  const float* x  = (const float*)d_in[0];
  const float* W0 = (const float*)d_in[1];  const float* b0 = (const float*)d_in[2];
  const float* W1 = (const float*)d_in[3];  const float* b1 = (const float*)d_in[4];
  const float* W2 = (const float*)d_in[5];  const float* b2 = (const float*)d_in[6];
  const float* W3 = (const float*)d_in[7];  const float* b3 = (const float*)d_in[8];
  const float* W4 = (const float*)d_in[9];  const float* b4 = (const float*)d_in[10];
  const float* Wr = (const float*)d_in[11]; const float* br = (const float*)d_in[12];
  const int* e0r = (const int*)d_in[13]; const int* e0c = (const int*)d_in[14];
  const int* e1r = (const int*)d_in[15]; const int* e1c = (const int*)d_in[16];
  const int* e2r = (const int*)d_in[17]; const int* e2c = (const int*)d_in[18];
  const int* e3r = (const int*)d_in[19]; const int* e3c = (const int*)d_in[20];
  const int* d0r = (const int*)d_in[21]; const int* d0c = (const int*)d_in[22];
  const float* d0v = (const float*)d_in[23];
  const int* d1r = (const int*)d_in[24]; const int* d1c = (const int*)d_in[25];
  const float* d1v = (const float*)d_in[26];
  const int* u3r = (const int*)d_in[27]; const int* u3c = (const int*)d_in[28];
  const float* u3v = (const float*)d_in[29];
  const int* u2r = (const int*)d_in[30]; const int* u2c = (const int*)d_in[31];
  const float* u2v = (const float*)d_in[32];

  size_t off = 0;
  auto alloc = [&](size_t nf) {
    nf = (nf + 3) & ~(size_t)3;        // keep every slab 16B aligned
    float* p = (float*)d_ws + off;
    off += nf;
    return p;
  };
  float* dinv0 = alloc(N0_);
  float* dinv1 = alloc(N0_);
  float* dinv2 = alloc(N1_);
  float* dinv3 = alloc(N2_);
  float* we0a  = alloc(E0_);   // e0 graph, lambda = 2.3
  float* we0b  = alloc(E0_);   // e0 graph, lambda = 2.0 (recon conv)
  float* we1   = alloc(E0_);
  float* we2   = alloc(E2_);
  float* we3   = alloc(E3_);
  float* wpack = alloc(192 * 16);
  float* bufA  = alloc((size_t)N0_ * B_ * 16);
  float* bufB  = alloc((size_t)N0_ * B_ * 16);
  float* Tst   = alloc((size_t)N0_ * B_ * 48);

  auto nb = [](int n) { return (unsigned)((n + TPB - 1) / TPB); };

  const float c23 = -2.f / 2.3f, diag23 = 2.f / 2.3f - 1.f;
  const float c20 = -1.f,        diag20 = 0.f;

  // ---- graph normalization ----
  k_zero<<<nb(N0_), TPB, 0, stream>>>(dinv0, N0_);
  k_zero<<<nb(N0_), TPB, 0, stream>>>(dinv1, N0_);
  k_zero<<<nb(N1_), TPB, 0, stream>>>(dinv2, N1_);
  k_zero<<<nb(N2_), TPB, 0, stream>>>(dinv3, N2_);
  k_deg<<<nb(E0_), TPB, 0, stream>>>(e0r, E0_, dinv0);
  k_deg<<<nb(E0_), TPB, 0, stream>>>(e1r, E0_, dinv1);
  k_deg<<<nb(E2_), TPB, 0, stream>>>(e2r, E2_, dinv2);
  k_deg<<<nb(E3_), TPB, 0, stream>>>(e3r, E3_, dinv3);
  k_dinv<<<nb(N0_), TPB, 0, stream>>>(dinv0, N0_);
  k_dinv<<<nb(N0_), TPB, 0, stream>>>(dinv1, N0_);
  k_dinv<<<nb(N1_), TPB, 0, stream>>>(dinv2, N1_);
  k_dinv<<<nb(N2_), TPB, 0, stream>>>(dinv3, N2_);
  k_edgew<<<nb(E0_), TPB, 0, stream>>>(e0r, e0c, dinv0, c23, we0a, E0_);
  k_edgew<<<nb(E0_), TPB, 0, stream>>>(e0r, e0c, dinv0, c20, we0b, E0_);
  k_edgew<<<nb(E0_), TPB, 0, stream>>>(e1r, e1c, dinv1, c23, we1, E0_);
  k_edgew<<<nb(E2_), TPB, 0, stream>>>(e2r, e2c, dinv2, c23, we2, E2_);
  k_edgew<<<nb(E3_), TPB, 0, stream>>>(e3r, e3c, dinv3, c23, we3, E3_);

  // generic weighted scatter dispatch: vector path when F % 4 == 0
  auto scatter = [&](const int* srow, const int* scol, const float* sw, float scale,
                     const float* src, float* dst, int srcStride, int dstStride,
                     int F, int E) {
    if ((F & 3) == 0) {
      int F4 = F >> 2;
      int tot = E * B_ * F4;
      k_scatter4<<<nb(tot), TPB, 0, stream>>>(srow, scol, sw, scale, src, dst,
                                              srcStride, dstStride, F4, tot);
    } else {
      int tot = E * B_ * F;
      k_scatter<<<nb(tot), TPB, 0, stream>>>(srow, scol, sw, scale, src, dst,
                                             srcStride, dstStride, F, tot);
    }
  };

  auto cheb = [&](const float* xin, bool xinIsBatchInput, int n, int Fin,
                  const int* grow, const int* gcol, const float* gw, int E, float diag,
                  const float* Wsrc, const float* bsrc, int Fout,
                  float* outp, int applyElu, int batchMajorN, int outStride) {
    int K = 6 * Fin;
    int rows = n * B_;
    int totF = rows * Fin;
    if (xinIsBatchInput)
      k_t0_from_x<<<nb(totF), TPB, 0, stream>>>(xin, Tst, n, Fin, K, totF);
    else
      k_t0_copy<<<nb(totF), TPB, 0, stream>>>(xin, Tst, Fin, K, totF);
    // T1 = diag*T0 + scatter(w * T0[col])
    k_lincomb<<<nb(totF), TPB, 0, stream>>>(Tst, K, Fin, 0, diag, 0, 0.f, Fin, totF);
    scatter(grow, gcol, gw, 1.f, Tst, Tst + Fin, K, K, Fin, E);
    for (int k = 2; k < 6; k++) {
      // Tk = 2*Lx(T_{k-1}) - T_{k-2}
      k_lincomb<<<nb(totF), TPB, 0, stream>>>(Tst, K, k * Fin, (k - 1) * Fin, 2.f * diag,
                                              (k - 2) * Fin, -1.f, Fin, totF);
      scatter(grow, gcol, gw, 2.f, Tst + (k - 1) * Fin, Tst + k * Fin, K, K, Fin, E);
    }
    k_packw<<<nb(K * 16), TPB, 0, stream>>>(Wsrc, wpack, K, Fout);
    int tiles = rows / 16;
    unsigned gblocks = (unsigned)((tiles + (TPB / 32) - 1) / (TPB / 32));
    k_gemm_wmma<<<gblocks, TPB, 0, stream>>>(Tst, K, wpack, bsrc, outp, rows, Fout,
                                             outStride, applyElu, batchMajorN);
  };

  auto pool = [&](const float* src, const int* prow, const int* pcol, const float* pval,
                  int E, int n_out, int F, float* outp) {
    int tot = n_out * B_ * F;
    k_zero<<<nb(tot), TPB, 0, stream>>>(outp, tot);
    scatter(prow, pcol, pval, 1.f, src, outp, F, F, F, E);
  };

  // encoder
  cheb(x, true, N0_, 6, e0r, e0c, we0a, E0_, diag23, W0, b0, 8, bufA, 1, 0, 8);
  cheb(bufA, false, N0_, 8, e1r, e1c, we1, E0_, diag23, W1, b1, 16, bufB, 1, 0, 16);
  pool(bufB, d0r, d0c, d0v, ND0_, N1_, 16, bufA);
  cheb(bufA, false, N1_, 16, e2r, e2c, we2, E2_, diag23, W2, b2, 32, bufB, 1, 0, 32);
  pool(bufB, d1r, d1c, d1v, ND1_, N2_, 32, bufA);
  // decoder
  cheb(bufA, false, N2_, 32, e3r, e3c, we3, E3_, diag23, W3, b3, 16, bufB, 1, 0, 16);
  pool(bufB, u3r, u3c, u3v, NU3_, N1_, 16, bufA);
  cheb(bufA, false, N1_, 16, e2r, e2c, we2, E2_, diag23, W4, b4, 8, bufB, 1, 0, 8);
  pool(bufB, u2r, u2c, u2v, NU2_, N0_, 8, bufA);
  // recon conv: lambda = 2.0 (diag = 0), writes [B, N0, 3] batch-major to d_out
  cheb(bufA, false, N0_, 8, e0r, e0c, we0b, E0_, diag20, Wr, br, 3,
       (float*)d_out, 0, N0_, 3);
}